// SeqSelfAttention_56667798504129
// MI455X (gfx1250) — compile-verified
//
#include <hip/hip_runtime.h>
#include <hip/hip_bf16.h>
#include <math.h>

typedef float v2f __attribute__((ext_vector_type(2)));
typedef float v8f __attribute__((ext_vector_type(8)));

#define T_DIM 1024
#define D_DIM 1024
#define U_DIM 64
#define BANDW 64
#define SWIN  80   // union of band over a 16-row t-tile: 16 + 64
#define QPAD 68    // qs row stride (conflict-free: 68 % 64 == 4)
#define EPAD 84    // es row stride (conflict-free fragment reads: 84 % 64 == 20)

// ---------------------------------------------------------------------------
// Async global -> LDS copy of 16 bytes per lane (CDNA5 async path, ASYNCcnt).
// LDS operand is the low 32 bits of the generic pointer (aperture truncation).
// ---------------------------------------------------------------------------
__device__ __forceinline__ void async_copy_b128(const float* gsrc, float* lds_dst) {
  unsigned ldsoff = (unsigned)(size_t)lds_dst;
  asm volatile("global_load_async_to_lds_b128 %0, %1, off"
               :: "v"(ldsoff), "v"(gsrc) : "memory");
}

// ---------------------------------------------------------------------------
// Kernel 1: q = x @ Wt, k = x @ Wx   (rows = B*T = 4096, K = D = 1024, N = 64)
// grid: 256 blocks (16 rows each), block: 128 threads = 4 waves (one per
// 16-wide u tile). Each wave produces a 16x16 tile of BOTH q and k sharing
// the same A fragment.
// ---------------------------------------------------------------------------
__global__ void __launch_bounds__(128)
proj_qk_kernel(const float* __restrict__ x, const float* __restrict__ Wt,
               const float* __restrict__ Wx, float* __restrict__ q,
               float* __restrict__ k) {
  const int row0 = blockIdx.x * 16;
  const int wave = threadIdx.x >> 5;
  const int lane = threadIdx.x & 31;
  const int u0   = wave * 16;          // U_DIM / 16 = 4 waves
  const int m    = lane & 15;          // A row / B col / C col
  const int kh   = (lane >> 4) * 2;    // K sub-offset: 0 for lanes 0-15, 2 for 16-31

  v8f accq = {};
  v8f acck = {};

  const float* __restrict__ xrow = x + (size_t)(row0 + m) * D_DIM;

  for (int kk = 0; kk < D_DIM; kk += 4) {
    // A fragment: 16x4 tile of x
    v2f a;
    a.x = xrow[kk + kh];
    a.y = xrow[kk + kh + 1];
    // B fragments: 4x16 tiles of Wt / Wx
    v2f bq, bk;
    bq.x = Wt[(size_t)(kk + kh) * U_DIM + u0 + m];
    bq.y = Wt[(size_t)(kk + kh + 1) * U_DIM + u0 + m];
    bk.x = Wx[(size_t)(kk + kh) * U_DIM + u0 + m];
    bk.y = Wx[(size_t)(kk + kh + 1) * U_DIM + u0 + m];

    accq = __builtin_amdgcn_wmma_f32_16x16x4_f32(false, a, false, bq,
                                                 (short)0, accq, false, false);
    acck = __builtin_amdgcn_wmma_f32_16x16x4_f32(false, a, false, bk,
                                                 (short)0, acck, false, false);
  }

  // C/D layout: VGPR i holds row M=i (lanes 0-15) or M=i+8 (lanes 16-31)
  const int rbase = row0 + ((lane >> 4) * 8);
#pragma unroll
  for (int i = 0; i < 8; ++i) {
    q[(size_t)(rbase + i) * U_DIM + u0 + m] = accq[i];
    k[(size_t)(rbase + i) * U_DIM + u0 + m] = acck[i];
  }
}

// ---------------------------------------------------------------------------
// Kernel 2: banded additive attention + output GEMM for one (b, 16-row tile).
// grid: 256 blocks, block: 256 threads = 8 waves.
// Output GEMM double-buffers x panels (80x16 f32) in LDS per wave via the
// CDNA5 async-to-LDS path, overlapping HBM/L2 fetch with WMMA issue.
// ---------------------------------------------------------------------------
__global__ void __launch_bounds__(256)
band_attn_kernel(const float* __restrict__ x, const float* __restrict__ q,
                 const float* __restrict__ k, const float* __restrict__ bh,
                 const float* __restrict__ Wa, const float* __restrict__ ba,
                 float* __restrict__ out) {
  const int tid  = threadIdx.x;
  const int bt   = blockIdx.x;          // 0..255
  const int b    = bt >> 6;             // 64 t-tiles per batch
  const int t0   = (bt & 63) * 16;
  const int s0   = t0 - BANDW / 2;      // window start (may be negative)

  __shared__ float qs[16][QPAD];        // padded: conflict-free column reads
  __shared__ float ks[SWIN][U_DIM];     // 20 KB (row-broadcast reads: no conflict)
  __shared__ float bhs[U_DIM];
  __shared__ float was[U_DIM];
  __shared__ float es[16][EPAD];        // scores -> probabilities, padded
  __shared__ float xstage[8][2][SWIN * 16];  // per-wave double-buffered x panel, 80 KB

  // ---- stage q tile, k window, bh, Wa into LDS ----
  for (int idx = tid; idx < 16 * U_DIM; idx += 256) {
    const int t = idx >> 6, u = idx & 63;
    qs[t][u] = q[(size_t)(b * T_DIM + t0 + t) * U_DIM + u];
  }
  for (int idx = tid; idx < SWIN * U_DIM; idx += 256) {
    const int si = idx >> 6, u = idx & 63;
    const int s = s0 + si;
    ks[si][u] = (s >= 0 && s < T_DIM)
                    ? k[(size_t)(b * T_DIM + s) * U_DIM + u] : 0.0f;
  }
  if (tid < U_DIM) {
    bhs[tid] = bh[tid];
    was[tid] = Wa[tid];
  }
  __syncthreads();

  const float bias_a = ba[0];

  // ---- band scores: e[t][si] = Wa . tanh(q_t + k_s + bh) + ba ----
  for (int p = tid; p < 16 * SWIN; p += 256) {
    const int t  = p & 15;
    const int si = p >> 4;               // 0..79
    const int s  = s0 + si;
    const int tg = t0 + t;
    float e = -1e30f;
    if (s >= 0 && s < T_DIM && s >= tg - BANDW / 2 && s < tg + BANDW / 2) {
      float acc = bias_a;
#pragma unroll 8
      for (int u = 0; u < U_DIM; ++u)
        acc = fmaf(tanhf(qs[t][u] + ks[si][u] + bhs[u]), was[u], acc);
      e = acc;
    }
    es[t][si] = e;
  }
  __syncthreads();

  // ---- per-row softmax over the 80-wide window ----
  if (tid < 16) {
    float mx = -1e30f;
#pragma unroll 8
    for (int si = 0; si < SWIN; ++si) mx = fmaxf(mx, es[tid][si]);
    float sum = 0.0f;
#pragma unroll 8
    for (int si = 0; si < SWIN; ++si) {
      const float p = __expf(es[tid][si] - mx);   // masked -> exp(-huge) = 0
      es[tid][si] = p;
      sum += p;
    }
    const float inv = 1.0f / sum;
#pragma unroll 8
    for (int si = 0; si < SWIN; ++si) es[tid][si] *= inv;
  }
  __syncthreads();

  // ---- output GEMM: v[16 x 1024] = a[16 x 80] @ x[s0..s0+80, 1024] ----
  const int wave = tid >> 5;
  const int lane = tid & 31;
  const int m    = lane & 15;
  const int kh   = (lane >> 4) * 2;
  const int rsh  = (lane >> 4) * 8;     // C row offset for this lane half
  const float* __restrict__ xb = x + (size_t)b * T_DIM * D_DIM;

  // Async fill of one 80x16 panel: 10 x b128, 8 rows (4 lanes/row) per op.
  const int frow  = lane >> 2;          // 0..7
  const int fcol4 = (lane & 3) * 4;     // 0,4,8,12
  float* mybuf0 = &xstage[wave][0][0];
  float* mybuf1 = &xstage[wave][1][0];

  auto fill_panel = [&](int d0, float* buf) {
#pragma unroll
    for (int i = 0; i < SWIN / 8; ++i) {
      const int row = frow + 8 * i;
      int s = s0 + row;
      s = s < 0 ? 0 : (s >= T_DIM ? T_DIM - 1 : s);   // clamped rows have a==0
      async_copy_b128(xb + (size_t)s * D_DIM + d0 + fcol4,
                      buf + row * 16 + fcol4);
    }
  };

  // prologue: fill buffer 0 for this wave's first d-tile
  fill_panel(wave * 16, mybuf0);

  int cur = 0;
  for (int nt = wave; nt < D_DIM / 16; nt += 8) {
    const int d0 = nt * 16;
    float* rbuf = cur ? mybuf1 : mybuf0;
    float* wbuf = cur ? mybuf0 : mybuf1;
    const bool have_next = (nt + 8) < D_DIM / 16;

    if (have_next) {
      fill_panel((nt + 8) * 16, wbuf);                 // overlap fetch w/ compute
      asm volatile("s_wait_asynccnt 0xa" ::: "memory"); // 10 in flight (next panel)
    } else {
      asm volatile("s_wait_asynccnt 0x0" ::: "memory");
    }

    v8f acc = {};
#pragma unroll
    for (int kk = 0; kk < SWIN; kk += 4) {
      v2f a;                               // A: probabilities (M=t, K=si)
      a.x = es[m][kk + kh];
      a.y = es[m][kk + kh + 1];
      v2f bf;                              // B: staged x panel (K=s row, N=d col)
      bf.x = rbuf[(kk + kh) * 16 + m];
      bf.y = rbuf[(kk + kh + 1) * 16 + m];
      acc = __builtin_amdgcn_wmma_f32_16x16x4_f32(false, a, false, bf,
                                                  (short)0, acc, false, false);
    }
#pragma unroll
    for (int i = 0; i < 8; ++i) {
      out[(size_t)(b * T_DIM + t0 + rsh + i) * D_DIM + d0 + m] = acc[i];
    }
    cur ^= 1;
  }
}

// ---------------------------------------------------------------------------
extern "C" void kernel_launch(void* const* d_in, const int* in_sizes, int n_in,
                              void* d_out, int out_size, void* d_ws, size_t ws_size,
                              hipStream_t stream) {
  const float* x  = (const float*)d_in[0];  // [4,1024,1024]
  const float* Wt = (const float*)d_in[1];  // [1024,64]
  const float* Wx = (const float*)d_in[2];  // [1024,64]
  const float* bh = (const float*)d_in[3];  // [64]
  const float* Wa = (const float*)d_in[4];  // [64,1]
  const float* ba = (const float*)d_in[5];  // [1]
  float* out = (float*)d_out;               // [4,1024,1024]

  float* q = (float*)d_ws;                  // [4096,64] = 1 MB
  float* k = q + (size_t)4 * T_DIM * U_DIM; // [4096,64] = 1 MB

  // 4096 rows / 16 = 256 tiles; 4 waves per block (one per 16-wide u tile)
  proj_qk_kernel<<<256, 128, 0, stream>>>(x, Wt, Wx, q, k);
  // 256 (b, t-tile) blocks, 8 waves each
  band_attn_kernel<<<256, 256, 0, stream>>>(x, q, k, bh, Wa, ba, out);
}